// HeteroRGCNLayer_86165633892368
// MI455X (gfx1250) — compile-verified
//
#include <hip/hip_runtime.h>
#include <hip/hip_bf16.h>

typedef __attribute__((ext_vector_type(2))) float v2f;
typedef __attribute__((ext_vector_type(8))) float v8f;

#define N_USER 50000
#define N_ITEM 50000
#define N_EDGE 1600000
#define D 64

// ---------------------------------------------------------------------------
// Zero-init (atomically-accumulated buffers must start at 0 on every call)
// ---------------------------------------------------------------------------
__global__ void zero_f32_kernel(float* __restrict__ p, int n) {
    int i = blockIdx.x * blockDim.x + threadIdx.x;
    int stride = gridDim.x * blockDim.x;
    for (; i < n; i += stride) p[i] = 0.0f;
}

// ---------------------------------------------------------------------------
// Dual-output fp32 WMMA GEMM:
//   Out1 = X @ W1 + b1   (per-etype linear -> workspace)
//   Out2 = X @ W2 + b2   (self-loop linear -> d_out base)
// X: [16*nTiles x 64], W: [64 x 64] row-major.
// W1 and W2 (16 KB each) are staged into LDS once per block with the CDNA5
// async global->LDS copy (ASYNCcnt-tracked), then the K-loop feeds WMMA
// B-fragments from DS loads instead of redundant VMEM loads.
// V_WMMA_F32_16X16X4_F32 fragment layout (ISA 7.12.2):
//   A 16x4:  lane l<16 holds row l, K=k0+0/1 in vgpr0/1; lanes 16-31 K=k0+2/3
//   B 4x16:  vgpr0: K=k0(+2 for hi half), N=lane&15 ; vgpr1: K=k0+1(+2)
//   C 16x16: vgpr r: M=r (lo half) / r+8 (hi half), N=lane&15
// ---------------------------------------------------------------------------
__global__ __launch_bounds__(128) void gemm_dual_wmma_kernel(
    const float* __restrict__ X,
    const float* __restrict__ W1, const float* __restrict__ b1, float* __restrict__ Out1,
    const float* __restrict__ W2, const float* __restrict__ b2, float* __restrict__ Out2,
    int nTiles) {
    __shared__ float sW[2 * D * D];   // [0..4095] = W1, [4096..8191] = W2, 32 KB

    const int tid  = threadIdx.x;
    const int lane = threadIdx.x & 31;
    const int wave = (blockIdx.x * blockDim.x + threadIdx.x) >> 5;

    // ---- Stage W1|W2 into LDS: 2048 float4 chunks, 16 per thread, async. ----
    {
        const unsigned lds_base = (unsigned)(uintptr_t)&sW[0];
#pragma unroll
        for (int i = 0; i < 16; ++i) {
            const int c = tid + 128 * i;            // chunk id, 0..2047
            const float* gsrc = (c < 1024) ? (W1 + (size_t)c * 4)
                                           : (W2 + (size_t)(c - 1024) * 4);
            const unsigned dlds = lds_base + (unsigned)c * 16u;
            asm volatile("global_load_async_to_lds_b128 %0, %1, off"
                         :: "v"(dlds), "v"(gsrc) : "memory");
        }
        asm volatile("s_wait_asynccnt 0" ::: "memory");
    }
    __syncthreads();                   // all waves' async copies are in LDS

    if (wave >= nTiles) return;        // wave-uniform: EXEC all-1s below
    const int row0 = wave << 4;
    const int hl   = lane >> 4;        // lane half: selects K sub-pair / M+8
    const int lm   = lane & 15;

    v8f c1[4], c2[4];
#pragma unroll
    for (int t = 0; t < 4; ++t) {
        const float bv1 = b1[t * 16 + lm];
        const float bv2 = b2[t * 16 + lm];
#pragma unroll
        for (int r = 0; r < 8; ++r) { c1[t][r] = bv1; c2[t][r] = bv2; }
    }

    const float* __restrict__ xrow = X + (size_t)(row0 + lm) * D;
    const float* __restrict__ sW1  = &sW[0];
    const float* __restrict__ sW2  = &sW[D * D];

#pragma unroll 4
    for (int k0 = 0; k0 < D; k0 += 4) {
        const int kb = k0 + hl * 2;
        v2f a;
        a.x = xrow[kb];
        a.y = xrow[kb + 1];
#pragma unroll
        for (int t = 0; t < 4; ++t) {
            const int col = t * 16 + lm;
            v2f bw1, bw2;
            bw1.x = sW1[kb * D + col];
            bw1.y = sW1[(kb + 1) * D + col];
            bw2.x = sW2[kb * D + col];
            bw2.y = sW2[(kb + 1) * D + col];
            c1[t] = __builtin_amdgcn_wmma_f32_16x16x4_f32(
                false, a, false, bw1, (short)0, c1[t], false, false);
            c2[t] = __builtin_amdgcn_wmma_f32_16x16x4_f32(
                false, a, false, bw2, (short)0, c2[t], false, false);
        }
    }

#pragma unroll
    for (int t = 0; t < 4; ++t) {
        const int col = t * 16 + lm;
#pragma unroll
        for (int r = 0; r < 8; ++r) {
            const int m = r + hl * 8;
            Out1[(size_t)(row0 + m) * D + col] = c1[t][r];
            Out2[(size_t)(row0 + m) * D + col] = c2[t][r];
        }
    }
}

// ---------------------------------------------------------------------------
// Edge scatter: agg[dst] += Wh[src], cnt[dst] += 1. One thread per
// (edge, feature) element -> coalesced 256B gather/scatter per edge;
// both tables are L2-resident (12.8 MB each vs 192 MB L2).
// ---------------------------------------------------------------------------
__global__ void scatter_sum_kernel(const float* __restrict__ Wh,
                                   const int* __restrict__ src,
                                   const int* __restrict__ dst,
                                   float* __restrict__ agg,
                                   float* __restrict__ cnt,
                                   int nEdges) {
    const int total  = nEdges * D;
    const int stride = gridDim.x * blockDim.x;
    for (int i = blockIdx.x * blockDim.x + threadIdx.x; i < total; i += stride) {
        const int e = i >> 6;
        const int f = i & (D - 1);
        const int s = src[e];
        const int d = dst[e];
        atomicAdd(&agg[(size_t)d * D + f], Wh[(size_t)s * D + f]);
        if (f == 0) atomicAdd(&cnt[d], 1.0f);
    }
}

// ---------------------------------------------------------------------------
// Finalize: out += agg / max(cnt, 1)   (out already holds X@loop_w + h_bias)
// ---------------------------------------------------------------------------
__global__ void finalize_kernel(const float* __restrict__ agg,
                                const float* __restrict__ cnt,
                                float* __restrict__ out, int nNodes) {
    const int total  = nNodes * D;
    const int stride = gridDim.x * blockDim.x;
    for (int i = blockIdx.x * blockDim.x + threadIdx.x; i < total; i += stride) {
        const int n = i >> 6;
        out[i] += agg[i] / fmaxf(cnt[n], 1.0f);
    }
}

extern "C" void kernel_launch(void* const* d_in, const int* in_sizes, int n_in,
                              void* d_out, int out_size, void* d_ws, size_t ws_size,
                              hipStream_t stream) {
    const float* user_feat = (const float*)d_in[0];
    const float* item_feat = (const float*)d_in[1];
    const int*   rate_src  = (const int*)d_in[2];
    const int*   rate_dst  = (const int*)d_in[3];
    const int*   rev_src   = (const int*)d_in[4];
    const int*   rev_dst   = (const int*)d_in[5];
    const float* W_rate    = (const float*)d_in[6];
    const float* b_rate    = (const float*)d_in[7];
    const float* W_rev     = (const float*)d_in[8];
    const float* b_rev     = (const float*)d_in[9];
    const float* loop_w    = (const float*)d_in[10];
    const float* h_bias    = (const float*)d_in[11];

    float* h_user = (float*)d_out;                       // [N_USER x 64]
    float* h_item = h_user + (size_t)N_USER * D;         // [N_ITEM x 64]

    float* Wh_rate  = (float*)d_ws;                      // [N_USER x 64]
    float* Wh_rev   = Wh_rate + (size_t)N_USER * D;      // [N_ITEM x 64]
    float* agg_item = Wh_rev  + (size_t)N_ITEM * D;      // [N_ITEM x 64]
    float* agg_user = agg_item + (size_t)N_ITEM * D;     // [N_USER x 64]
    float* cnt_item = agg_user + (size_t)N_USER * D;     // [N_ITEM]
    float* cnt_user = cnt_item + N_ITEM;                 // [N_USER]

    // agg_item..cnt_user is one contiguous region; zero it every call.
    const int ztot = N_ITEM * D + N_USER * D + N_ITEM + N_USER;
    zero_f32_kernel<<<4096, 256, 0, stream>>>(agg_item, ztot);

    // GEMM phase: per-etype linear -> ws, self-loop linear + bias -> d_out.
    const int nTilesU = N_USER / 16;   // 3125 (exact)
    const int nTilesI = N_ITEM / 16;   // 3125 (exact)
    gemm_dual_wmma_kernel<<<(nTilesU + 3) / 4, 128, 0, stream>>>(
        user_feat, W_rate, b_rate, Wh_rate, loop_w, h_bias, h_user, nTilesU);
    gemm_dual_wmma_kernel<<<(nTilesI + 3) / 4, 128, 0, stream>>>(
        item_feat, W_rev, b_rev, Wh_rev, loop_w, h_bias, h_item, nTilesI);

    // Edge scatter phase (L2-resident atomics).
    scatter_sum_kernel<<<32768, 256, 0, stream>>>(
        Wh_rate, rate_src, rate_dst, agg_item, cnt_item, N_EDGE);
    scatter_sum_kernel<<<32768, 256, 0, stream>>>(
        Wh_rev, rev_src, rev_dst, agg_user, cnt_user, N_EDGE);

    // Mean + add into self-loop result.
    finalize_kernel<<<8192, 256, 0, stream>>>(agg_item, cnt_item, h_item, N_ITEM);
    finalize_kernel<<<8192, 256, 0, stream>>>(agg_user, cnt_user, h_user, N_USER);
}